// BWCaster_53283364274861
// MI455X (gfx1250) — compile-verified
//
#include <hip/hip_runtime.h>
#include <hip/hip_bf16.h>

// ---------------------------------------------------------------------------
// BWCaster forward for MI455X (gfx1250, wave32).
//  J=24 joints, C=16 channels, G=128 grid, N=R*S=131072 points.
//  - Transpose planes/lines to channel-innermost layout (vectorized L2 gathers)
//  - Per-point tri-plane sampling -> normalized LBS weights wn^T [24][N]
//  - LBS blend M = T^T x wn^T via V_WMMA_F32_16X16X4_F32 (K=24 in 6 steps)
//  - xw = M*[p,1]; vw = M[:3,:3]*q  (algebraic simplification of xw-warp(p-q))
// ---------------------------------------------------------------------------

typedef float v2f __attribute__((ext_vector_type(2)));
typedef float v8f __attribute__((ext_vector_type(8)));

constexpr int J = 24;
constexpr int C = 16;
constexpr int G = 128;
constexpr int NPTS = 2048 * 64;           // 131072
constexpr int PLANE_T_ELEMS = 3 * J * G * G * C;   // 18,874,368
constexpr int LINE_T_ELEMS  = 3 * J * G * C;       // 147,456
constexpr int WN_ELEMS      = J * NPTS;            // 3,145,728

// ---------------------------------------------------------------------------
// Transpose [J,C,G,G] -> [k][J][G][G][C]  (channel-innermost)
// ---------------------------------------------------------------------------
__global__ __launch_bounds__(256) void transpose_planes_kernel(
    const float* __restrict__ p0, const float* __restrict__ p1,
    const float* __restrict__ p2, float* __restrict__ outT)
{
    for (size_t i = (size_t)blockIdx.x * blockDim.x + threadIdx.x;
         i < (size_t)PLANE_T_ELEMS; i += (size_t)gridDim.x * blockDim.x) {
        int c = (int)(i & 15);
        size_t r = i >> 4;
        int x = (int)(r & 127); r >>= 7;
        int y = (int)(r & 127); r >>= 7;
        int j = (int)(r % J);
        int k = (int)(r / J);
        const float* src = (k == 0) ? p0 : (k == 1) ? p1 : p2;
        outT[i] = src[(((size_t)j * C + c) * G + y) * G + x];
    }
}

// Transpose [J,C,G] -> [k][J][G][C]
__global__ __launch_bounds__(256) void transpose_lines_kernel(
    const float* __restrict__ l0, const float* __restrict__ l1,
    const float* __restrict__ l2, float* __restrict__ outT)
{
    for (size_t i = (size_t)blockIdx.x * blockDim.x + threadIdx.x;
         i < (size_t)LINE_T_ELEMS; i += (size_t)gridDim.x * blockDim.x) {
        int c = (int)(i & 15);
        size_t r = i >> 4;
        int z = (int)(r & 127); r >>= 7;
        int j = (int)(r % J);
        int k = (int)(r / J);
        const float* src = (k == 0) ? l0 : (k == 1) ? l1 : l2;
        outT[i] = src[((size_t)j * C + c) * G + z];
    }
}

// ---------------------------------------------------------------------------
// Bilinear plane x line sample, channel-innermost layout.
// plane: [G][G][C], line: [G][C]. Returns sum over 16 channels.
// ---------------------------------------------------------------------------
__device__ __forceinline__ float sample_mode_T(
    const float* __restrict__ plane, const float* __restrict__ line,
    float xn, float yn, float zn)
{
    const float A = 0.5f * (float)(G - 1);
    float fx = (xn + 1.f) * A, fy = (yn + 1.f) * A, fz = (zn + 1.f) * A;
    float x0f = floorf(fx), y0f = floorf(fy), z0f = floorf(fz);
    float wx1 = fx - x0f, wx0 = 1.f - wx1;
    float wy1 = fy - y0f, wy0 = 1.f - wy1;
    float wz1 = fz - z0f, wz0 = 1.f - wz1;
    int ix0 = (int)x0f, iy0 = (int)y0f, iz0 = (int)z0f;
    int ix1 = ix0 + 1,  iy1 = iy0 + 1,  iz1 = iz0 + 1;
    float vx0 = (ix0 >= 0 && ix0 < G) ? 1.f : 0.f;
    float vx1 = (ix1 >= 0 && ix1 < G) ? 1.f : 0.f;
    float vy0 = (iy0 >= 0 && iy0 < G) ? 1.f : 0.f;
    float vy1 = (iy1 >= 0 && iy1 < G) ? 1.f : 0.f;
    float vz0 = (iz0 >= 0 && iz0 < G) ? 1.f : 0.f;
    float vz1 = (iz1 >= 0 && iz1 < G) ? 1.f : 0.f;
    int cx0 = min(max(ix0, 0), G - 1), cx1 = min(max(ix1, 0), G - 1);
    int cy0 = min(max(iy0, 0), G - 1), cy1 = min(max(iy1, 0), G - 1);
    int cz0 = min(max(iz0, 0), G - 1), cz1 = min(max(iz1, 0), G - 1);

    // line factor per channel: lf[c] = a0*L0[c] + a1*L1[c]
    const float a0 = wz0 * vz0, a1 = wz1 * vz1;
    const float4* L0 = (const float4*)(line + (size_t)cz0 * C);
    const float4* L1 = (const float4*)(line + (size_t)cz1 * C);
    float4 lf[4];
#pragma unroll
    for (int t = 0; t < 4; ++t) {
        float4 u = L0[t], v = L1[t];
        lf[t].x = a0 * u.x + a1 * v.x;
        lf[t].y = a0 * u.y + a1 * v.y;
        lf[t].z = a0 * u.z + a1 * v.z;
        lf[t].w = a0 * u.w + a1 * v.w;
    }

    const float w00 = wx0 * wy0 * vx0 * vy0;
    const float w01 = wx1 * wy0 * vx1 * vy0;
    const float w10 = wx0 * wy1 * vx0 * vy1;
    const float w11 = wx1 * wy1 * vx1 * vy1;

    float s = 0.f;
    const int corn_x[4] = {cx0, cx1, cx0, cx1};
    const int corn_y[4] = {cy0, cy0, cy1, cy1};
    const float corn_w[4] = {w00, w01, w10, w11};
#pragma unroll
    for (int q = 0; q < 4; ++q) {
        const float4* P = (const float4*)(plane + ((size_t)corn_y[q] * G + corn_x[q]) * C);
        float d = 0.f;
#pragma unroll
        for (int t = 0; t < 4; ++t) {
            float4 pv = P[t];
            d += pv.x * lf[t].x + pv.y * lf[t].y + pv.z * lf[t].z + pv.w * lf[t].w;
        }
        s += corn_w[q] * d;
    }
    return s;
}

// Fallback sampler for original [C][G][G] / [C][G] layout (small workspace).
__device__ __forceinline__ float sample_mode_orig(
    const float* __restrict__ plane, const float* __restrict__ line,
    float xn, float yn, float zn)
{
    const float A = 0.5f * (float)(G - 1);
    float fx = (xn + 1.f) * A, fy = (yn + 1.f) * A, fz = (zn + 1.f) * A;
    float x0f = floorf(fx), y0f = floorf(fy), z0f = floorf(fz);
    float wx1 = fx - x0f, wx0 = 1.f - wx1;
    float wy1 = fy - y0f, wy0 = 1.f - wy1;
    float wz1 = fz - z0f, wz0 = 1.f - wz1;
    int ix0 = (int)x0f, iy0 = (int)y0f, iz0 = (int)z0f;
    int ix1 = ix0 + 1,  iy1 = iy0 + 1,  iz1 = iz0 + 1;
    float vx0 = (ix0 >= 0 && ix0 < G) ? 1.f : 0.f;
    float vx1 = (ix1 >= 0 && ix1 < G) ? 1.f : 0.f;
    float vy0 = (iy0 >= 0 && iy0 < G) ? 1.f : 0.f;
    float vy1 = (iy1 >= 0 && iy1 < G) ? 1.f : 0.f;
    float vz0 = (iz0 >= 0 && iz0 < G) ? 1.f : 0.f;
    float vz1 = (iz1 >= 0 && iz1 < G) ? 1.f : 0.f;
    int cx0 = min(max(ix0, 0), G - 1), cx1 = min(max(ix1, 0), G - 1);
    int cy0 = min(max(iy0, 0), G - 1), cy1 = min(max(iy1, 0), G - 1);
    int cz0 = min(max(iz0, 0), G - 1), cz1 = min(max(iz1, 0), G - 1);
    const float w00 = wx0 * wy0 * vx0 * vy0;
    const float w01 = wx1 * wy0 * vx1 * vy0;
    const float w10 = wx0 * wy1 * vx0 * vy1;
    const float w11 = wx1 * wy1 * vx1 * vy1;
    const float a0 = wz0 * vz0, a1 = wz1 * vz1;
    float s = 0.f;
#pragma unroll
    for (int c = 0; c < C; ++c) {
        const float* pc = plane + (size_t)c * G * G;
        float pv = w00 * pc[(size_t)cy0 * G + cx0] + w01 * pc[(size_t)cy0 * G + cx1]
                 + w10 * pc[(size_t)cy1 * G + cx0] + w11 * pc[(size_t)cy1 * G + cx1];
        float lv = a0 * line[(size_t)c * G + cz0] + a1 * line[(size_t)c * G + cz1];
        s += pv * lv;
    }
    return s;
}

// ---------------------------------------------------------------------------
// Per-point: local coords per joint, tri-plane sigma, relu + normalize,
// store wn^T [J][N] (joint-major so the WMMA B-operand loads are coalesced).
// ---------------------------------------------------------------------------
__global__ __launch_bounds__(256) void weights_kernel(
    const float* __restrict__ xyz, const float* __restrict__ Tm,
    const float* __restrict__ aabb,
    const float* __restrict__ p0, const float* __restrict__ p1, const float* __restrict__ p2,
    const float* __restrict__ l0, const float* __restrict__ l1, const float* __restrict__ l2,
    const float* __restrict__ planesT, const float* __restrict__ linesT,
    float* __restrict__ wnT, int useT)
{
    __shared__ float sT[J * 16];
    for (int i = threadIdx.x; i < J * 16; i += blockDim.x) sT[i] = Tm[i];
    __syncthreads();

    const int n = blockIdx.x * blockDim.x + threadIdx.x;   // exact cover of NPTS
    const float px = xyz[(size_t)n * 3 + 0];
    const float py = xyz[(size_t)n * 3 + 1];
    const float pz = xyz[(size_t)n * 3 + 2];

    float sc[3], bs[3];
#pragma unroll
    for (int a = 0; a < 3; ++a) {
        const float lo = aabb[a], hi = aabb[3 + a];
        sc[a] = 2.f / (hi - lo);
        bs[a] = -lo * sc[a] - 1.f;
    }

    const int m0[3] = {0, 0, 1}, m1[3] = {1, 2, 2}, vm[3] = {2, 1, 0};
    const float* planes_orig[3] = {p0, p1, p2};
    const float* lines_orig[3]  = {l0, l1, l2};

    float w[J];
    float wsum = 0.f;
#pragma unroll 1
    for (int j = 0; j < J; ++j) {
        const float* tj = sT + j * 16;
        float nc[3];
#pragma unroll
        for (int a = 0; a < 3; ++a) {
            float la = tj[a * 4 + 0] * px + tj[a * 4 + 1] * py
                     + tj[a * 4 + 2] * pz + tj[a * 4 + 3];
            nc[a] = la * sc[a] + bs[a];
        }
        float sigma = 0.f;
#pragma unroll
        for (int k = 0; k < 3; ++k) {
            const float xn = nc[m0[k]], yn = nc[m1[k]], zn = nc[vm[k]];
            if (useT) {
                const float* plane = planesT + ((size_t)k * J + j) * (size_t)G * G * C;
                const float* line  = linesT  + ((size_t)k * J + j) * (size_t)G * C;
                sigma += sample_mode_T(plane, line, xn, yn, zn);
            } else {
                const float* plane = planes_orig[k] + (size_t)j * C * G * G;
                const float* line  = lines_orig[k]  + (size_t)j * C * G;
                sigma += sample_mode_orig(plane, line, xn, yn, zn);
            }
        }
        float wj = sigma > 0.f ? sigma : 0.f;   // relu
        w[j] = wj;
        wsum += wj;
    }
    const float inv = 1.f / (wsum + 1e-6f);
#pragma unroll
    for (int j = 0; j < J; ++j)
        wnT[(size_t)j * NPTS + n] = w[j] * inv;
}

// ---------------------------------------------------------------------------
// LBS blend via V_WMMA_F32_16X16X4_F32:
//   D(16x16) = A(16x4) x B(4x16) accumulated over 6 K-chunks (J=24).
//   A = T^T chunk:  A[e][k] = Tm[(j0+k)*16 + e]   (e = flattened 4x4 element)
//   B = wn^T chunk: B[k][n] = wnT[(j0+k)*NPTS + n0 + n]
// D layout (32-bit C/D 16x16): VGPR v, lanes 0-15 -> M=v, N=lane;
//   lanes 16-31 -> M=v+8, N=lane-16.  So lane l (<16) holds matrix rows 0,1
//   of point n0+l, lane l+16 holds rows 2,3 of the same point.
// ---------------------------------------------------------------------------
__global__ __launch_bounds__(256) void blend_kernel(
    const float* __restrict__ xyz, const float* __restrict__ vdir,
    const float* __restrict__ Tm, const float* __restrict__ wnT,
    float* __restrict__ out)
{
    const int lane = threadIdx.x & 31;
    const int wave = threadIdx.x >> 5;
    const int tile = blockIdx.x * 8 + wave;       // 8192 tiles of 16 points
    const int n0 = tile * 16;
    const int half = lane >> 4;                   // 0: K=0,1 rows / 1: K=2,3
    const int l = lane & 15;

    v8f acc = {0.f, 0.f, 0.f, 0.f, 0.f, 0.f, 0.f, 0.f};
#pragma unroll
    for (int kk = 0; kk < 6; ++kk) {
        const int j0 = kk * 4;
        v2f a, b;
        // A: 16x4 f32 layout — lanes 0-15: M=l, VGPR0=K0,VGPR1=K1;
        //                      lanes 16-31: M=l, VGPR0=K2,VGPR1=K3.
        a.x = Tm[(j0 + 2 * half + 0) * 16 + l];
        a.y = Tm[(j0 + 2 * half + 1) * 16 + l];
        // B: 4x16 f32 (rows striped across lanes, K split across lane halves)
        b.x = wnT[(size_t)(j0 + 2 * half + 0) * NPTS + n0 + l];
        b.y = wnT[(size_t)(j0 + 2 * half + 1) * NPTS + n0 + l];
        acc = __builtin_amdgcn_wmma_f32_16x16x4_f32(
            false, a, false, b, (short)0, acc, false, false);
    }

    const int pn = n0 + l;
    const float px = xyz[(size_t)pn * 3 + 0];
    const float py = xyz[(size_t)pn * 3 + 1];
    const float pz = xyz[(size_t)pn * 3 + 2];
    const float qx = vdir[(size_t)pn * 3 + 0];
    const float qy = vdir[(size_t)pn * 3 + 1];
    const float qz = vdir[(size_t)pn * 3 + 2];

    // acc[0..3] = matrix row (2*half), acc[4..7] = row (2*half+1), row-major 4x4
    const float r0 = acc[0] * px + acc[1] * py + acc[2] * pz + acc[3];
    const float r1 = acc[4] * px + acc[5] * py + acc[6] * pz + acc[7];
    const float v0 = acc[0] * qx + acc[1] * qy + acc[2] * qz;
    const float v1 = acc[4] * qx + acc[5] * qy + acc[6] * qz;

    float* xw = out;
    float* vw = out + (size_t)NPTS * 3;
    if (half == 0) {           // rows 0,1 -> x,y components
        xw[(size_t)pn * 3 + 0] = r0;
        xw[(size_t)pn * 3 + 1] = r1;
        vw[(size_t)pn * 3 + 0] = v0;
        vw[(size_t)pn * 3 + 1] = v1;
    } else {                   // row 2 -> z component (row 3 discarded)
        xw[(size_t)pn * 3 + 2] = r0;
        vw[(size_t)pn * 3 + 2] = v0;
    }
}

// ---------------------------------------------------------------------------
extern "C" void kernel_launch(void* const* d_in, const int* in_sizes, int n_in,
                              void* d_out, int out_size, void* d_ws, size_t ws_size,
                              hipStream_t stream) {
    const float* xyz  = (const float*)d_in[0];
    const float* vdir = (const float*)d_in[1];
    const float* Tm   = (const float*)d_in[2];
    const float* p0   = (const float*)d_in[3];
    const float* p1   = (const float*)d_in[4];
    const float* p2   = (const float*)d_in[5];
    const float* l0   = (const float*)d_in[6];
    const float* l1   = (const float*)d_in[7];
    const float* l2   = (const float*)d_in[8];
    const float* aabb = (const float*)d_in[9];
    float* out = (float*)d_out;

    float* wnT     = (float*)d_ws;             // J*NPTS floats
    float* planesT = wnT + WN_ELEMS;           // 3*J*G*G*C floats
    float* linesT  = planesT + PLANE_T_ELEMS;  // 3*J*G*C floats
    const size_t need = (size_t)(WN_ELEMS + PLANE_T_ELEMS + LINE_T_ELEMS) * sizeof(float);
    const int useT = (ws_size >= need) ? 1 : 0;

    if (useT) {
        transpose_planes_kernel<<<8192, 256, 0, stream>>>(p0, p1, p2, planesT);
        transpose_lines_kernel<<<576, 256, 0, stream>>>(l0, l1, l2, linesT);
    }

    weights_kernel<<<NPTS / 256, 256, 0, stream>>>(
        xyz, Tm, aabb, p0, p1, p2, l0, l1, l2, planesT, linesT, wnT, useT);

    // 131072 points / 16 per wave / 8 waves per block = 1024 blocks
    blend_kernel<<<NPTS / 128, 256, 0, stream>>>(xyz, vdir, Tm, wnT, out);
}